// ClasswiseECELoss_32195074850952
// MI455X (gfx1250) — compile-verified
//
#include <hip/hip_runtime.h>
#include <stdint.h>

#define NBINS 15
#define NCLS  256
#define SEGS  (NCLS * NBINS)      // 3840
#define BLOCK 256
#define WPB   8                   // waves per block (wave32)
#define NBUF  4                   // per-wave row buffers (prefetch distance 3)
#define GRID  1024

__global__ void zero_ws_kernel(float* ws, int n) {
  int i = blockIdx.x * blockDim.x + threadIdx.x;
  if (i < n) ws[i] = 0.0f;
}

// Issue one 1KB row copy (2 x 512B async b128, NT on the global side since
// logits are streamed exactly once; keeps L2 free for the atomic accumulators).
#define ISSUE_ROW(row, ldsaddr)                                                \
  do {                                                                         \
    unsigned voff = (unsigned)(row) * 1024u + (unsigned)lane * 16u;            \
    asm volatile(                                                              \
        "global_load_async_to_lds_b128 %0, %1, %2 offset:0 th:TH_LOAD_NT\n\t"  \
        "global_load_async_to_lds_b128 %0, %1, %2 offset:512 th:TH_LOAD_NT"    \
        :: "v"(ldsaddr), "v"(voff), "s"(logits) : "memory");                   \
  } while (0)

__global__ __launch_bounds__(BLOCK)
void ece_hist_kernel(const float* __restrict__ logits,
                     const int* __restrict__ labels,
                     float* __restrict__ g_conf,
                     float* __restrict__ g_corr,
                     int N) {
  __shared__ float h_conf[SEGS];                 // 15 KB
  __shared__ float h_corr[SEGS];                 // 15 KB
  __shared__ float rowbuf[WPB][NBUF][NCLS];      // 32 KB, 4-deep per-wave pipeline

  const int tid = threadIdx.x;
  for (int i = tid; i < SEGS; i += BLOCK) { h_conf[i] = 0.0f; h_corr[i] = 0.0f; }
  __syncthreads();

  const int wave   = tid >> 5;
  const int lane   = tid & 31;
  const int gwave  = blockIdx.x * WPB + wave;
  const int stride = gridDim.x * WPB;

  // Per-lane LDS byte address of this wave's buffer 0; buffer q = +q*1024.
  const unsigned ldsbase =
      (unsigned)(uintptr_t)(&rowbuf[wave][0][0]) + (unsigned)lane * 16u;

  // Prologue: stage up to 3 rows ahead.
  #pragma unroll
  for (int k = 0; k < NBUF - 1; ++k) {
    int rk = gwave + k * stride;
    if (rk < N) ISSUE_ROW(rk, ldsbase + (unsigned)k * 1024u);
  }

  int p = 0;
  for (int r = gwave; r < N; r += stride) {
    const int rp = r + (NBUF - 1) * stride;
    if (rp < N) ISSUE_ROW(rp, ldsbase + (unsigned)((p + NBUF - 1) & (NBUF - 1)) * 1024u);

    // Copies complete in order: current buffer is ready once only the
    // (2 * rows-ahead) most recent async ops may still be outstanding.
    const int ahead = (r + stride < N) + (r + 2 * stride < N) + (r + 3 * stride < N);
    if (ahead == 3)      asm volatile("s_wait_asynccnt 6" ::: "memory");
    else if (ahead == 2) asm volatile("s_wait_asynccnt 4" ::: "memory");
    else if (ahead == 1) asm volatile("s_wait_asynccnt 2" ::: "memory");
    else                 asm volatile("s_wait_asynccnt 0" ::: "memory");

    const float* buf = &rowbuf[wave][p][0];

    // Lane L owns classes L + 32j (15L mod 64 unique => conflict-free atomics).
    float x[8];
    #pragma unroll
    for (int j = 0; j < 8; ++j) x[j] = buf[lane + 32 * j];

    // Row max: 8 local + 5 wave32 shuffle steps.
    float m = x[0];
    #pragma unroll
    for (int j = 1; j < 8; ++j) m = fmaxf(m, x[j]);
    #pragma unroll
    for (int d = 16; d >= 1; d >>= 1) m = fmaxf(m, __shfl_xor(m, d, 32));

    // exp + row sum.
    float e[8];
    float s = 0.0f;
    #pragma unroll
    for (int j = 0; j < 8; ++j) {
      e[j] = exp2f((x[j] - m) * 1.4426950408889634f);
      s += e[j];
    }
    #pragma unroll
    for (int d = 16; d >= 1; d >>= 1) s += __shfl_xor(s, d, 32);
    const float inv = __builtin_amdgcn_rcpf(s);

    const int label = labels[r];

    #pragma unroll
    for (int j = 0; j < 8; ++j) {
      const float conf = e[j] * inv;
      // searchsorted(uppers, conf, 'left') == ceil(conf*15)-1 clipped to [0,14]
      int b = (int)ceilf(conf * 15.0f) - 1;
      b = b < 0 ? 0 : (b > NBINS - 1 ? NBINS - 1 : b);
      const int c = lane + 32 * j;
      atomicAdd(&h_conf[c * NBINS + b], conf);          // ds_add_f32
      if (c == label) atomicAdd(&h_corr[c * NBINS + b], 1.0f);
    }
    p = (p + 1) & (NBUF - 1);
  }

  __syncthreads();
  // Flush block-private histogram to global accumulators (L2-resident, 30 KB).
  for (int i = tid; i < SEGS; i += BLOCK) {
    atomicAdd(&g_conf[i], h_conf[i]);
    atomicAdd(&g_corr[i], h_corr[i]);
  }
}

__global__ __launch_bounds__(256)
void ece_final_kernel(const float* __restrict__ g_conf,
                      const float* __restrict__ g_corr,
                      float* __restrict__ out, int N) {
  __shared__ float red[256];
  const int tid = threadIdx.x;
  float acc = 0.0f;
  for (int i = tid; i < SEGS; i += 256) acc += fabsf(g_conf[i] - g_corr[i]);
  red[tid] = acc;
  __syncthreads();
  for (int s = 128; s > 0; s >>= 1) {
    if (tid < s) red[tid] += red[tid + s];
    __syncthreads();
  }
  if (tid == 0) out[0] = red[0] / ((float)N * (float)NCLS);
}

extern "C" void kernel_launch(void* const* d_in, const int* in_sizes, int n_in,
                              void* d_out, int out_size, void* d_ws, size_t ws_size,
                              hipStream_t stream) {
  const float* logits = (const float*)d_in[0];
  const int*   labels = (const int*)d_in[1];
  const int N = in_sizes[1];               // labels count = number of rows

  float* g_conf = (float*)d_ws;
  float* g_corr = g_conf + SEGS;

  // Workspace is poisoned by the harness and never re-zeroed between replays:
  // zero the accumulators on every call.
  zero_ws_kernel<<<(2 * SEGS + 255) / 256, 256, 0, stream>>>(g_conf, 2 * SEGS);

  ece_hist_kernel<<<GRID, BLOCK, 0, stream>>>(logits, labels, g_conf, g_corr, N);

  ece_final_kernel<<<1, 256, 0, stream>>>(g_conf, g_corr, (float*)d_out, N);
}